// SimpleSTGCN_36447092474005
// MI455X (gfx1250) — compile-verified
//
#include <hip/hip_runtime.h>
#include <math.h>

// Problem dims (fixed by the reference)
#define B_  16
#define T_  12
#define N_  2000
#define C_  2
#define H_  64
#define E_  16000
#define BN_ (N_*B_)      // 32000
#define EN_ (E_+N_)      // 18000
#define G_  256          // 4*H (gates)
#define KC_ 128          // concat K = H(x) + H(h)
#define WT_PITCH 130     // LDS pitch for K=128 rows (stride-2-bank padding)

typedef __attribute__((ext_vector_type(2))) float v2f;
typedef __attribute__((ext_vector_type(8))) float v8f;

// Branch-free fast activations: v_exp_f32 + v_rcp_f32 (TRANS ops, co-execute
// with the matrix pipe; no EXEC divergence in the WMMA hot loop).
__device__ __forceinline__ float sigf(float x){
  return __builtin_amdgcn_rcpf(1.0f + __expf(-x));
}
__device__ __forceinline__ float tanh_fast(float x){
  // 2*sigmoid(2x) - 1; saturates to +-1 correctly for large |x|
  return 2.0f * __builtin_amdgcn_rcpf(1.0f + __expf(-2.0f*x)) - 1.0f;
}

// ---------------------------------------------------------------- utilities
__global__ void fill_zero_f32(float* __restrict__ p, long n){
  long i = (long)blockIdx.x*blockDim.x + threadIdx.x;
  long s = (long)gridDim.x*blockDim.x;
  for (; i < n; i += s) p[i] = 0.0f;
}

// ---------------------------------------------------------------- GCN: norm
__global__ void deg_kernel(const int* __restrict__ ei, const float* __restrict__ ew,
                           float* __restrict__ deg){
  int i = blockIdx.x*blockDim.x + threadIdx.x;
  if (i < E_)       atomicAdd(&deg[ei[E_ + i]], ew[i]);    // dst-indexed degree
  else if (i < EN_) atomicAdd(&deg[i - E_], 1.0f);         // self loops
}

__global__ void norm_kernel(const int* __restrict__ ei, const float* __restrict__ ew,
                            const float* __restrict__ deg, float* __restrict__ nrm){
  int i = blockIdx.x*blockDim.x + threadIdx.x;
  if (i >= EN_) return;
  int s, d; float w;
  if (i < E_){ s = ei[i]; d = ei[E_ + i]; w = ew[i]; }
  else       { s = d = i - E_; w = 1.0f; }
  float degs = deg[s], degd = deg[d];
  float dis_s = degs > 0.0f ? 1.0f/sqrtf(degs) : 0.0f;
  float dis_d = degd > 0.0f ? 1.0f/sqrtf(degd) : 0.0f;
  nrm[i] = dis_s * w * dis_d;
}

// ------------------------------------------------ GCN: xw = x @ W  (C=2 -> trivial)
// xw layout: (N, B, T, H) so the per-node slab is contiguous for the edge gather.
__global__ void xw_kernel(const float* __restrict__ x, const float* __restrict__ gw,
                          float* __restrict__ xw){
  long idx = (long)blockIdx.x*blockDim.x + threadIdx.x;
  if (idx >= (long)N_*B_*T_*H_) return;
  int h = (int)(idx & (H_-1));
  long r = idx >> 6;
  int t = (int)(r % T_); r /= T_;
  int b = (int)(r % B_);
  int n = (int)(r / B_);
  const float* xp = x + (((long)(b*T_ + t))*N_ + n)*C_;   // x is (B,T,N,C)
  xw[idx] = xp[0]*gw[h] + xp[1]*gw[H_ + h];
}

// ------------------------------------------------ GCN: edge scatter-add
// agg layout = LSTM seq layout (T, BN, H) with bn = n*B + b.
__global__ void scatter_kernel(const int* __restrict__ ei, const float* __restrict__ nrm,
                               const float* __restrict__ xw, float* __restrict__ agg){
  int e = blockIdx.x;            // one block per (edge or self-loop)
  int s, d;
  if (e < E_){ s = ei[e]; d = ei[E_ + e]; } else { s = d = e - E_; }
  float w = nrm[e];
  if (w == 0.0f) return;         // block-uniform
  const float* src = xw + (long)s * (B_*T_*H_);
  for (int i = threadIdx.x; i < B_*T_*H_; i += blockDim.x){
    int b   = i / (T_*H_);
    int rem = i - b*(T_*H_);
    int t   = rem >> 6;
    int h   = rem & 63;
    atomicAdd(&agg[((long)t*BN_ + (d*B_ + b))*H_ + h], w * src[i]);
  }
}

__global__ void bias_relu_kernel(float* __restrict__ seq, const float* __restrict__ gb){
  long idx = (long)blockIdx.x*blockDim.x + threadIdx.x;
  if (idx >= (long)T_*BN_*H_) return;
  float v = seq[idx] + gb[idx & (H_-1)];
  seq[idx] = v > 0.0f ? v : 0.0f;
}

// ------------------------------------------------ LSTM weight prep
// wcat[g][k] : k<64 -> w_ih[g][k], k>=64 -> w_hh[g][k-64]; bcat = b_ih + b_hh
__global__ void prep_wcat(const float* __restrict__ wih, const float* __restrict__ whh,
                          const float* __restrict__ bih, const float* __restrict__ bhh,
                          float* __restrict__ wcat, float* __restrict__ bcat){
  int i = blockIdx.x*blockDim.x + threadIdx.x;
  if (i < G_*KC_){
    int n = i >> 7, k = i & 127;
    wcat[i] = (k < H_) ? wih[n*H_ + k] : whh[n*H_ + (k - H_)];
  } else if (i < G_*KC_ + G_){
    int j = i - G_*KC_;
    bcat[j] = bih[j] + bhh[j];
  }
}

// ------------------------------------------------ fused whole-layer LSTM
// One block = 8 waves = 128 rows; loops t=0..11 internally. Weights (256x128 f32)
// live in LDS for the whole layer; h/c never leave the block. Each wave computes,
// per 16-column group jt, the i/f/g/o 16x16 D-tiles (tiles jt, jt+4, jt+8, jt+12)
// with V_WMMA_F32_16X16X4_F32 chained over K=128 (32 chunks), then does the gate
// nonlinearity lane-locally on the D layout (M = v + 8*hi, N = lane%16).
__global__ __launch_bounds__(256, 1) void lstm_layer_kernel(
    const float* __restrict__ xin,     // (T, BN, H) layer input
    float* __restrict__ hout,          // (T, BN, H) hidden states out
    const float* __restrict__ wcat,    // (256, 128)
    const float* __restrict__ bcat)    // (256)
{
  __shared__ float s_wt[G_ * WT_PITCH];        // 133,120 B
  __shared__ float s_a [8 * 16 * WT_PITCH];    //  66,560 B  (per-wave [x|h] tiles)

  const int tid  = threadIdx.x;
  const int wave = tid >> 5;                   // wave32
  const int lane = tid & 31;
  const int lrow = lane & 15;
  const int hi   = lane >> 4;                  // K/M-half select

  // stage weights: s_wt[g][k], padded pitch
  for (int i = tid; i < G_*KC_; i += 256){
    int n = i >> 7, k = i & 127;
    s_wt[n*WT_PITCH + k] = wcat[i];
  }
  float* aw = s_a + wave * (16*WT_PITCH);
  for (int i = lane; i < 16*H_; i += 32){      // h(-1) = 0
    int r = i >> 6, c = i & 63;
    aw[r*WT_PITCH + H_ + c] = 0.0f;
  }
  __syncthreads();

  const int row0 = blockIdx.x * 128 + wave * 16;

  float breg[4][4];
  #pragma unroll
  for (int jt = 0; jt < 4; ++jt)
    #pragma unroll
    for (int g = 0; g < 4; ++g)
      breg[jt][g] = bcat[g*H_ + jt*16 + lrow];

  float creg[4][8];
  #pragma unroll
  for (int jt = 0; jt < 4; ++jt)
    #pragma unroll
    for (int v = 0; v < 8; ++v)
      creg[jt][v] = 0.0f;

  #pragma unroll 1
  for (int t = 0; t < T_; ++t){
    // stage this wave's 16x64 x_t tile (coalesced)
    const float* xt = xin + ((long)t*BN_ + row0) * H_;
    for (int i = lane; i < 16*H_; i += 32){
      int r = i >> 6, c = i & 63;
      aw[r*WT_PITCH + c] = xt[i];
    }

    float hreg[4][8];
    #pragma unroll
    for (int jt = 0; jt < 4; ++jt){
      float bi = breg[jt][0], bf = breg[jt][1], bg = breg[jt][2], bo = breg[jt][3];
      v8f ai = {bi,bi,bi,bi,bi,bi,bi,bi};
      v8f af = {bf,bf,bf,bf,bf,bf,bf,bf};
      v8f ag = {bg,bg,bg,bg,bg,bg,bg,bg};
      v8f ao = {bo,bo,bo,bo,bo,bo,bo,bo};
      const float* wrow = s_wt + (jt*16 + lrow)*WT_PITCH + 2*hi;
      const float* arow = aw + lrow*WT_PITCH + 2*hi;
      #pragma unroll 4
      for (int kc = 0; kc < 32; ++kc){
        v2f a  = *(const v2f*)(arow + 4*kc);                       // A frag (16x4)
        v2f b0 = *(const v2f*)(wrow + 4*kc);                       // i tile
        v2f b1 = *(const v2f*)(wrow + 4*kc +  64*WT_PITCH);        // f tile
        v2f b2 = *(const v2f*)(wrow + 4*kc + 128*WT_PITCH);        // g tile
        v2f b3 = *(const v2f*)(wrow + 4*kc + 192*WT_PITCH);        // o tile
        ai = __builtin_amdgcn_wmma_f32_16x16x4_f32(false, a, false, b0, (short)0, ai, false, false);
        af = __builtin_amdgcn_wmma_f32_16x16x4_f32(false, a, false, b1, (short)0, af, false, false);
        ag = __builtin_amdgcn_wmma_f32_16x16x4_f32(false, a, false, b2, (short)0, ag, false, false);
        ao = __builtin_amdgcn_wmma_f32_16x16x4_f32(false, a, false, b3, (short)0, ao, false, false);
      }
      // lane-local LSTM gate update on the D layout (branch-free activations)
      #pragma unroll
      for (int v = 0; v < 8; ++v){
        float iv = sigf(ai[v]);
        float fv = sigf(af[v]);
        float gv = tanh_fast(ag[v]);
        float ov = sigf(ao[v]);
        float cv = fv*creg[jt][v] + iv*gv;
        creg[jt][v] = cv;
        hreg[jt][v] = ov * tanh_fast(cv);
      }
    }

    // write h: to global, and back into the A-staging tile (transpose via LDS)
    float* op = hout + (long)t*BN_*H_;
    #pragma unroll
    for (int jt = 0; jt < 4; ++jt){
      #pragma unroll
      for (int v = 0; v < 8; ++v){
        int rr = v + 8*hi;
        int j  = jt*16 + lrow;
        float hv = hreg[jt][v];
        aw[rr*WT_PITCH + H_ + j] = hv;
        op[(long)(row0 + rr)*H_ + j] = hv;
      }
    }
  }
}

// ---------------------------------------------------------------- final FC
__global__ void fc_kernel(const float* __restrict__ last, const float* __restrict__ fcw,
                          const float* __restrict__ fcb, float* __restrict__ out){
  int i = blockIdx.x*blockDim.x + threadIdx.x;
  if (i >= B_*N_) return;
  int b = i / N_, n = i - b*N_;
  const float* hp = last + (long)(n*B_ + b)*H_;
  float acc = fcb[0];
  #pragma unroll
  for (int k = 0; k < H_; ++k) acc += hp[k]*fcw[k];
  out[i] = acc;       // out is (B, N)
}

// ---------------------------------------------------------------- launcher
extern "C" void kernel_launch(void* const* d_in, const int* in_sizes, int n_in,
                              void* d_out, int out_size, void* d_ws, size_t ws_size,
                              hipStream_t stream) {
  (void)in_sizes; (void)n_in; (void)out_size; (void)ws_size;
  const float* x    = (const float*)d_in[0];
  const int*   ei   = (const int*)  d_in[1];
  const float* ew   = (const float*)d_in[2];
  const float* gcnw = (const float*)d_in[3];
  const float* gcnb = (const float*)d_in[4];
  const float* wih0 = (const float*)d_in[5];
  const float* whh0 = (const float*)d_in[6];
  const float* bih0 = (const float*)d_in[7];
  const float* bhh0 = (const float*)d_in[8];
  const float* wih1 = (const float*)d_in[9];
  const float* whh1 = (const float*)d_in[10];
  const float* bih1 = (const float*)d_in[11];
  const float* bhh1 = (const float*)d_in[12];
  const float* fcw  = (const float*)d_in[13];
  const float* fcb  = (const float*)d_in[14];
  float* out = (float*)d_out;

  // workspace carve-up (floats); total ~188 MB
  float* ws    = (float*)d_ws;
  float* deg   = ws;                       // 2000   (pad to 2048)
  float* nrm   = ws + 2048;                // 18000  (pad to 20480 total)
  float* bcat0 = ws + 20480;               // 256
  float* bcat1 = ws + 20736;               // 256
  float* wcat0 = ws + 20992;               // 32768
  float* wcat1 = ws + 53760;               // 32768
  float* xw    = ws + 86528;               // 24,576,000  (N,B,T,H); reused as seqA
  float* seq0  = xw + (long)N_*B_*T_*H_;   // 24,576,000  (T,BN,H)

  const long SEQ_ELEMS = (long)T_*BN_*H_;  // == N*B*T*H

  // 1) zero the aggregation target and degree
  fill_zero_f32<<<2048, 256, 0, stream>>>(seq0, SEQ_ELEMS);
  fill_zero_f32<<<8,    256, 0, stream>>>(deg, N_);

  // 2) GCN normalization
  deg_kernel <<<(EN_ + 255)/256, 256, 0, stream>>>(ei, ew, deg);
  norm_kernel<<<(EN_ + 255)/256, 256, 0, stream>>>(ei, ew, deg, nrm);

  // 3) xw = x @ gcn_w, node-major
  xw_kernel<<<(int)((SEQ_ELEMS + 255)/256), 256, 0, stream>>>(x, gcnw, xw);

  // 4) edge scatter-add into seq layout (xw is L2-resident: 98 MB < 192 MB)
  scatter_kernel<<<EN_, 256, 0, stream>>>(ei, nrm, xw, seq0);

  // 5) bias + ReLU
  bias_relu_kernel<<<(int)((SEQ_ELEMS + 255)/256), 256, 0, stream>>>(seq0, gcnb);

  // 6) LSTM weight prep
  prep_wcat<<<(G_*KC_ + G_ + 255)/256, 256, 0, stream>>>(wih0, whh0, bih0, bhh0, wcat0, bcat0);
  prep_wcat<<<(G_*KC_ + G_ + 255)/256, 256, 0, stream>>>(wih1, whh1, bih1, bhh1, wcat1, bcat1);

  // 7) two LSTM layers (ping-pong seq0 <-> xw buffer)
  lstm_layer_kernel<<<BN_/128, 256, 0, stream>>>(seq0, xw,   wcat0, bcat0);
  lstm_layer_kernel<<<BN_/128, 256, 0, stream>>>(xw,   seq0, wcat1, bcat1);

  // 8) FC on last timestep
  const float* last = seq0 + (long)(T_-1)*BN_*H_;
  fc_kernel<<<(B_*N_ + 255)/256, 256, 0, stream>>>(last, fcw, fcb, out);
}